// HOGLayerC_5437428597310
// MI455X (gfx1250) — compile-verified
//
#include <hip/hip_runtime.h>
#include <stdint.h>

// ---------------------------------------------------------------------------
// Fused HOG layer for MI455X (gfx1250, wave32):
//   reflect-pad -> depthwise 3x3 Sobel (gx,gy) -> norm/bin -> 7x7 sum-pool
//   -> per-cell L2 normalize over 9 bins.
// One workgroup per (b*3+c, cell_row). CDNA5 paths used:
//   * TENSOR_LOAD_TO_LDS (TDM) for the 9x224 input row band  (tensorcnt)
//   * V_WMMA_F32_16X16X4_F32 selection-matrix GEMM for the 7-col pooling
//   * ds_* LDS traffic for the fused intermediate (no 255MB dense one-hot)
// Memory roofline: ~49.5MB read (9/7 band overlap) + 7.1MB write ~= 2.4us
// at 23.3TB/s; compute is ~0.5 GFLOP -> firmly bandwidth-bound.
// ---------------------------------------------------------------------------

#define HOG_NBINS 9
#define HOG_W     224
#define HOG_H     224
#define HOG_CELLS 32
#define HOG_COLH_STRIDE 17   // 9 bins + zero pad, odd stride -> no bank conflicts

typedef float        v2f __attribute__((ext_vector_type(2)));
typedef float        v8f __attribute__((ext_vector_type(8)));
typedef unsigned int v4u __attribute__((ext_vector_type(4)));
typedef int          v4i __attribute__((ext_vector_type(4)));
typedef int          v8i __attribute__((ext_vector_type(8)));

#if defined(__has_builtin)
#  if __has_builtin(__builtin_amdgcn_tensor_load_to_lds)
#    define HOG_HAS_TDM 1
#  endif
#  if __has_builtin(__builtin_amdgcn_wmma_f32_16x16x4_f32)
#    define HOG_HAS_WMMA 1
#  endif
#endif
#ifndef HOG_HAS_TDM
#define HOG_HAS_TDM 0
#endif
#ifndef HOG_HAS_WMMA
#define HOG_HAS_WMMA 0
#endif

// ---- compile-time probes (show up in hipcc stderr, do not fail the build) --
#if HOG_HAS_TDM
#warning "HOG-PROBE: tensor_load_to_lds builtin PRESENT -> TDM path enabled"
#else
#warning "HOG-PROBE: tensor_load_to_lds builtin MISSING -> cooperative-load fallback"
#endif
#if HOG_HAS_WMMA
#warning "HOG-PROBE: wmma_f32_16x16x4_f32 builtin PRESENT"
#else
#warning "HOG-PROBE: wmma_f32_16x16x4_f32 builtin MISSING -> scalar pooling fallback"
#endif
#if defined(__has_builtin) && __has_builtin(__builtin_amdgcn_global_load_async_to_lds_b32)
#warning "HOG-PROBE: global_load_async_to_lds_b32 builtin PRESENT (unused, FYI)"
#else
#warning "HOG-PROBE: global_load_async_to_lds_b32 builtin MISSING"
#endif

#if HOG_HAS_TDM
#  if __has_include(<hip/amd_detail/amd_gfx1250_TDM.h>)
#    define HOG_TDM_6ARG 1   // therock-10.0 headers => 6-arg builtin
#warning "HOG-PROBE: using 6-arg tensor_load_to_lds form"
#  else
#    define HOG_TDM_6ARG 0   // ROCm 7.2 => 5-arg builtin
#warning "HOG-PROBE: using 5-arg tensor_load_to_lds form"
#  endif
#endif

// Sector boundaries cos/sin(k*pi/9), k = 1..8 (20-degree bins, half-plane folded)
__constant__ float HOG_BCOS[8] = {
    0.9396926208f, 0.7660444431f, 0.5000000000f, 0.1736481777f,
   -0.1736481777f, -0.5000000000f, -0.7660444431f, -0.9396926208f };
__constant__ float HOG_BSIN[8] = {
    0.3420201433f, 0.6427876097f, 0.8660254038f, 0.9848077530f,
    0.9848077530f, 0.8660254038f, 0.6427876097f, 0.3420201433f };

__global__ __launch_bounds__(256)
void hog_fused_kernel(const float* __restrict__ x,
                      const float* __restrict__ wxg,
                      const float* __restrict__ wyg,
                      float* __restrict__ out)
{
    __shared__ float tile[9 * HOG_W];                        // 9 rows x 224 cols
    __shared__ float haloL[9];                               // reflected col -1
    __shared__ float haloR[9];                               // reflected col 224
    __shared__ float colH[HOG_W * HOG_COLH_STRIDE];          // per-column 9-bin hist (7 rows summed)
    __shared__ float cellV[HOG_CELLS * HOG_NBINS];           // pooled cell histograms

    const int tid = threadIdx.x;
    const int cy  = blockIdx.x & 31;          // cell row 0..31
    const int bc  = blockIdx.x >> 5;          // b*3 + c
    const int ch  = bc % 3;
    const float* plane = x + (size_t)bc * (HOG_H * HOG_W);

    // Row band: global rows 7*cy-1 .. 7*cy+7 -> tile slots 0..8 (reflect at edges)
    int g_row0, nrows, slot0;
    if (cy == 0)       { g_row0 = 0;   nrows = 8; slot0 = 1; }   // slot0 filled by reflect copy
    else if (cy == 31) { g_row0 = 216; nrows = 8; slot0 = 0; }   // slot8 filled by reflect copy
    else               { g_row0 = 7 * cy - 1; nrows = 9; slot0 = 0; }

#if HOG_HAS_TDM
    if (tid == 0) {
        // 2D TDM descriptor per CDNA5 ISA 8.3/8.4: tile = nrows x 224 f32, stride 224.
        uint64_t ga  = (uint64_t)(uintptr_t)(plane + (size_t)g_row0 * HOG_W);
        uint32_t lds = (uint32_t)(uintptr_t)(void*)&tile[slot0 * HOG_W]; // low32 of generic ptr = LDS addr
        v4u g0; v8i g1; v4i g2 = (v4i)0; v4i g3 = (v4i)0;
        g0[0] = 1u;                                            // count=1, user descriptor
        g0[1] = lds;                                           // lds_addr (bytes)
        g0[2] = (uint32_t)ga;                                  // global_addr[31:0]
        g0[3] = (uint32_t)((ga >> 32) & 0x1FFFFFFu) | (2u << 30); // global_addr[56:32], type=2
        g1[0] = (2 << 16);                                     // data_size = 4B
        g1[1] = (HOG_W & 0xFFFF) << 16;                        // tensor_dim0 lo16
        g1[2] = ((nrows & 0xFFFF) << 16);                      // dim0 hi16=0 | tensor_dim1 lo16
        g1[3] = (HOG_W & 0xFFFF) << 16;                        // dim1 hi16=0 | tile_dim0 = 224
        g1[4] = (nrows & 0xFFFF);                              // tile_dim1 | tile_dim2=0
        g1[5] = HOG_W;                                         // tensor_dim0_stride lo32
        g1[6] = 0;                                             // stride0 hi16 | stride1 lo16
        g1[7] = 0;                                             // stride1 hi32
#if HOG_TDM_6ARG
        __builtin_amdgcn_tensor_load_to_lds(g0, g1, g2, g3, (v8i)0, 0);
#else
        __builtin_amdgcn_tensor_load_to_lds(g0, g1, g2, g3, 0);
#endif
    }
#else
    // Fallback: cooperative plain loads of the band
    for (int i = tid; i < nrows * HOG_W; i += 256)
        tile[slot0 * HOG_W + i] = plane[(size_t)g_row0 * HOG_W + i];
#endif

    // Overlap with TDM: zero the column histograms and fetch halo columns.
    for (int i = tid; i < HOG_W * HOG_COLH_STRIDE; i += 256) colH[i] = 0.0f;
    if (tid < 9) {
        int grow = 7 * cy - 1 + tid;
        if (grow < 0)   grow = 1;          // reflect of row -1
        if (grow > 223) grow = 222;        // reflect of row 224
        haloL[tid] = plane[(size_t)grow * HOG_W + 1];    // reflect of col -1
        haloR[tid] = plane[(size_t)grow * HOG_W + 222];  // reflect of col 224
    }

#if HOG_HAS_TDM
#  if defined(__has_builtin) && __has_builtin(__builtin_amdgcn_s_wait_tensorcnt)
    __builtin_amdgcn_s_wait_tensorcnt(0);   // no-op for non-issuing waves (TENSORcnt==0)
#  else
    asm volatile("s_wait_tensorcnt 0x0" ::: "memory");
#  endif
#endif
    __syncthreads();
    // Reflect-fill the missing edge row inside LDS
    if (cy == 0  && tid < HOG_W) tile[tid]             = tile[2 * HOG_W + tid];
    if (cy == 31 && tid < HOG_W) tile[8 * HOG_W + tid] = tile[6 * HOG_W + tid];
    __syncthreads();

    // Depthwise 3x3 weights for this channel (identical across threads -> cached)
    float wx[9], wy[9];
#pragma unroll
    for (int i = 0; i < 9; ++i) { wx[i] = wxg[ch * 9 + i]; wy[i] = wyg[ch * 9 + i]; }

    // Each of 224 threads owns one column; slide a 3x3 window down 7 output rows.
    if (tid < HOG_W) {
        const int col = tid;
        const bool lft = (col == 0), rgt = (col == HOG_W - 1);
        float l0 = lft ? haloL[0] : tile[col - 1];
        float c0 = tile[col];
        float r0 = rgt ? haloR[0] : tile[col + 1];
        float l1 = lft ? haloL[1] : tile[HOG_W + col - 1];
        float c1 = tile[HOG_W + col];
        float r1 = rgt ? haloR[1] : tile[HOG_W + col + 1];
#pragma unroll
        for (int rr = 1; rr <= 7; ++rr) {
            float l2 = lft ? haloL[rr + 1] : tile[(rr + 1) * HOG_W + col - 1];
            float c2 = tile[(rr + 1) * HOG_W + col];
            float r2 = rgt ? haloR[rr + 1] : tile[(rr + 1) * HOG_W + col + 1];
            float gx = wx[0]*l0 + wx[1]*c0 + wx[2]*r0
                     + wx[3]*l1 + wx[4]*c1 + wx[5]*r1
                     + wx[6]*l2 + wx[7]*c2 + wx[8]*r2;
            float gy = wy[0]*l0 + wy[1]*c0 + wy[2]*r0
                     + wy[3]*l1 + wy[4]*c1 + wy[5]*r1
                     + wy[6]*l2 + wy[7]*c2 + wy[8]*r2;
            float nrm = sqrtf(gx * gx + gy * gy);
            // bin = floor(atan2(gx,gy)*9/pi) mod 9 ; mod-pi symmetry -> fold to sin>=0
            bool flip = (gx < 0.0f) || ((gx == 0.0f) && (gy < 0.0f));
            float fx = flip ? -gx : gx;
            float fy = flip ? -gy : gy;
            int bin = 0;
#pragma unroll
            for (int k = 0; k < 8; ++k)
                bin += (fx * HOG_BCOS[k] > fy * HOG_BSIN[k]) ? 1 : 0;
            colH[col * HOG_COLH_STRIDE + bin] += nrm;   // exclusive column -> no atomic
            l0 = l1; c0 = c1; r0 = r1;
            l1 = l2; c1 = c2; r1 = r2;
        }
    }
    __syncthreads();

    // 7-wide column pooling as selection-matrix GEMM on the matrix pipe:
    //   cell(32x9) = S(32x224) * colH(224x9), S[m,k] = (k/7 == m).
    // Waves 0,1 each own a 16-cell M-tile; 56 chained V_WMMA_F32_16X16X4_F32.
    // k/7 is tracked incrementally (k advances by 4) to keep the VALU feeding
    // the accumulator chain to a couple of ops instead of a mul_hi divide.
#if HOG_HAS_WMMA
    if (tid < 64) {                    // waves 0,1 fully active -> EXEC all ones
        const int lane   = tid & 31;
        const int mt     = tid >> 5;           // M tile (cells 0-15 / 16-31)
        const int n      = lane & 15;          // bin lane (valid if < 9)
        const int hi     = lane >> 4;          // 0: K+0/1, 1: K+2/3 (A layout)
        const int mycell = mt * 16 + n;        // A-matrix row for this lane
        int kq   = 0;                          // (ks*4 + 2*hi) / 7, tracked incrementally
        int krm  = 2 * hi;                     // (ks*4 + 2*hi) % 7
        int boff = 2 * hi * HOG_COLH_STRIDE + n;   // LDS offset of b.x
        v8f acc = (v8f)0.0f;
#pragma unroll 4
        for (int ks = 0; ks < 56; ++ks) {
            v2f a, b;
            a.x = (kq == mycell) ? 1.0f : 0.0f;
            a.y = ((kq + ((krm == 6) ? 1 : 0)) == mycell) ? 1.0f : 0.0f;
            b.x = colH[boff];                       // pad lanes n>=9 read zeros
            b.y = colH[boff + HOG_COLH_STRIDE];
            acc = __builtin_amdgcn_wmma_f32_16x16x4_f32(
                      false, a, false, b, (short)0, acc, false, false);
            krm += 4;
            int carry = (krm >= 7) ? 1 : 0;
            kq  += carry;
            krm -= 7 * carry;
            boff += 4 * HOG_COLH_STRIDE;
        }
        if (n < HOG_NBINS) {
#pragma unroll
            for (int j = 0; j < 8; ++j) {
                int cell = mt * 16 + hi * 8 + j;     // D layout: vgpr j -> M=j / M=j+8
                cellV[cell * HOG_NBINS + n] = acc[j];
            }
        }
    }
#else
    if (tid < HOG_CELLS * HOG_NBINS) {
        int cell = tid / HOG_NBINS, bin = tid % HOG_NBINS;
        float s = 0.0f;
        for (int c7 = 0; c7 < 7; ++c7)
            s += colH[(cell * 7 + c7) * HOG_COLH_STRIDE + bin];
        cellV[tid] = s;
    }
#endif
    __syncthreads();

    // Per-cell L2 normalize over the 9 bins, coalesced store to [b,c,9,32,32]
    if (tid < HOG_CELLS) {
        float v[HOG_NBINS];
        float s = 0.0f;
#pragma unroll
        for (int bin = 0; bin < HOG_NBINS; ++bin) {
            v[bin] = cellV[tid * HOG_NBINS + bin];
            s += v[bin] * v[bin];
        }
        float d = fmaxf(sqrtf(s), 1e-12f);
        size_t obase = (size_t)bc * HOG_NBINS * (HOG_CELLS * HOG_CELLS)
                     + (size_t)cy * HOG_CELLS + tid;
#pragma unroll
        for (int bin = 0; bin < HOG_NBINS; ++bin)
            out[obase + (size_t)bin * (HOG_CELLS * HOG_CELLS)] = v[bin] / d;
    }
}

extern "C" void kernel_launch(void* const* d_in, const int* in_sizes, int n_in,
                              void* d_out, int out_size, void* d_ws, size_t ws_size,
                              hipStream_t stream) {
    (void)n_in; (void)out_size; (void)d_ws; (void)ws_size;
    const float* x  = (const float*)d_in[0];
    const float* wx = (const float*)d_in[1];
    const float* wy = (const float*)d_in[2];
    float* out = (float*)d_out;
    const int planes = in_sizes[0] / (HOG_W * HOG_H);   // B*3
    dim3 grid(planes * HOG_CELLS);
    hog_fused_kernel<<<grid, 256, 0, stream>>>(x, wx, wy, out);
}